// selfMarginLoss_90099823936143
// MI455X (gfx1250) — compile-verified
//
#include <hip/hip_runtime.h>

typedef __attribute__((ext_vector_type(2))) float v2f;
typedef __attribute__((ext_vector_type(8))) float v8f;

#define MARGIN 0.1f
#define WAVES_PER_BLOCK 8
#define MAX_N 64            // padded per-wave tile space (reference N = 50)
#define PAD_B (-1.0e30f)    // column pad: d = -1e30 + c < 0  -> relu kills j >= n
#define PAD_C ( 1.0e30f)    // row pad:    c = margin - 1e30  -> relu kills i >= n

// D = A x B + C with A[m,0]=1 (K=0 only): D[i,j] = ranked[j] - ranked[i] + MARGIN.
// NT > 0: compile-time tile count (full unroll). NT == 0: dynamic fallback.
template <int NT>
__device__ __forceinline__ float groupLoss(const float* __restrict__ rkB,
                                           const float* __restrict__ rkC,
                                           const float* __restrict__ wt,
                                           int lane, int ntDyn)
{
    const int nt     = (NT > 0) ? NT : ntDyn;
    const int rowOff = (lane >= 16) ? 8 : 0;   // C/D layout: M = vgpr + (lane<16 ? 0 : 8)
    const int jcol   = lane & 15;              // column index within tile
    const int tdm    = jcol - rowOff;          // diag mask: keep iff tdm > r (ti-invariant)

    v2f a;                                     // rank-1: A[m,0] = 1, all other K zero
    a.x = (lane < 16) ? 1.0f : 0.0f;
    a.y = 0.0f;

    float acc = 0.0f;

    #pragma unroll
    for (int ti = 0; ti < nt; ++ti) {
        const int i0 = ti << 4;

        v8f c;                                 // consecutive -> ds_load_b128 + v_pk_add_f32
        #pragma unroll
        for (int r = 0; r < 8; ++r)
            c[r] = MARGIN - rkC[i0 + r + rowOff];

        v8f rowAcc;

        // ---- diagonal tile: only the ti-invariant j>i mask; validity via padding ----
        {
            v2f b;                             // K!=0 lane slots multiply A-zeros: no select
            b.x = rkB[i0 + jcol];
            b.y = 0.0f;

            v8f d = __builtin_amdgcn_wmma_f32_16x16x4_f32(
                false, a, false, b, (short)0, c, false, false);

            #pragma unroll
            for (int r = 0; r < 8; ++r) {
                const float sel = (tdm > r) ? d[r] : -1.0f;
                rowAcc[r] = fmaxf(sel, 0.0f);
            }
        }

        // ---- off-diagonal tiles: j>i guaranteed; padding handles all validity ----
        #pragma unroll
        for (int tj = ti + 1; tj < nt; ++tj) {
            v2f b;
            b.x = rkB[(tj << 4) + jcol];
            b.y = 0.0f;

            v8f d = __builtin_amdgcn_wmma_f32_16x16x4_f32(
                false, a, false, b, (short)0, c, false, false);

            #pragma unroll
            for (int r = 0; r < 8; ++r)
                rowAcc[r] += fmaxf(d[r], 0.0f);
        }

        // mean-reduction row weights, once per ti (consecutive -> b128 loads)
        #pragma unroll
        for (int r = 0; r < 8; ++r)
            acc += rowAcc[r] * wt[i0 + r + rowOff];
    }
    return acc;
}

__global__ __launch_bounds__(256) void selfMarginLoss_wmma_kernel(
    const float* __restrict__ scores,
    const int*   __restrict__ werRank,
    float*       __restrict__ out,
    int numGroups, int n)
{
    __shared__ float rankedB[WAVES_PER_BLOCK][MAX_N];  // -1e30 padded (column source)
    __shared__ float rankedC[WAVES_PER_BLOCK][MAX_N];  // +1e30 padded (row source)
    __shared__ float wtab[MAX_N];                      // 1 / max(n-1-i, 1)
    __shared__ float blockAcc;

    const int lane = threadIdx.x & 31;
    const int wave = threadIdx.x >> 5;
    const int g    = blockIdx.x * WAVES_PER_BLOCK + wave;
    const bool active = (g < numGroups);

    if (threadIdx.x == 0) blockAcc = 0.0f;
    if (threadIdx.x < MAX_N) {
        int den = n - 1 - (int)threadIdx.x;
        wtab[threadIdx.x] = 1.0f / (float)(den < 1 ? 1 : den);
    }

    rankedB[wave][lane]      = PAD_B;
    rankedB[wave][lane + 32] = PAD_B;
    rankedC[wave][lane]      = PAD_C;
    rankedC[wave][lane + 32] = PAD_C;

    if (active) {
        const long long base = (long long)g * n;
        for (int r = lane; r < n; r += 32) {
            const int rkidx = werRank[base + r];        // permutation within group
            const float v   = scores[base + rkidx];     // gather ranked score
            rankedB[wave][r] = v;
            rankedC[wave][r] = v;
        }
    }
    __syncthreads();

    float acc = 0.0f;
    if (active) {
        const int nt = (n + 15) >> 4;                   // 4 for n = 50
        acc = (nt == 4)
            ? groupLoss<4>(rankedB[wave], rankedC[wave], wtab, lane, nt)
            : groupLoss<0>(rankedB[wave], rankedC[wave], wtab, lane,
                           (nt > 4 ? 4 : nt));          // MAX_N = 64 cap
    }

    // wave32 butterfly reduction (warpSize == 32 on gfx1250)
    #pragma unroll
    for (int m = 16; m > 0; m >>= 1)
        acc += __shfl_xor(acc, m, 32);

    if (lane == 0)
        atomicAdd(&blockAcc, acc);      // ds_add_f32
    __syncthreads();
    if (threadIdx.x == 0)
        atomicAdd(out, blockAcc);       // one global_atomic_add_f32 per block
}

extern "C" void kernel_launch(void* const* d_in, const int* in_sizes, int n_in,
                              void* d_out, int out_size, void* d_ws, size_t ws_size,
                              hipStream_t stream) {
    const float* scores  = (const float*)d_in[0];
    // d_in[1] = nBestIndex (uniform group sizes; N derived from shapes)
    const int*   werRank = (const int*)d_in[2];

    const int B = in_sizes[1];              // number of groups
    const int N = in_sizes[0] / B;          // n-best size (50)

    // Zero the scalar accumulator each call (graph-capture safe).
    hipMemsetAsync(d_out, 0, sizeof(float), stream);

    const int blocks = (B + WAVES_PER_BLOCK - 1) / WAVES_PER_BLOCK;
    selfMarginLoss_wmma_kernel<<<blocks, 256, 0, stream>>>(
        scores, werRank, (float*)d_out, B, N);
}